// VSSBlock_Cross_new_80564996538478
// MI455X (gfx1250) — compile-verified
//
#include <hip/hip_runtime.h>
#include <hip/hip_bf16.h>
#include <math.h>

#define BB 4
#define DD 192
#define HH 96
#define WW 96
#define KK 4
#define NN 16
#define RR 12
#define W2 48
#define LL 2304          // (96/2)*(96/2)
#define CPROJ 44         // R + 2N
#define CPAD 48          // padded to 3x16 rows
#define BKTOT 16         // B*K
#define NCH 3072         // B*K*D channels
#define CHUNKS 36
#define CLEN 64          // 36*64 = 2304
#define TSUB 16          // LDS sub-tile steps
#define LDP 17           // padded LDS stride (bank-conflict free)

typedef __attribute__((ext_vector_type(2))) float v2f;
typedef __attribute__((ext_vector_type(8))) float v8f;

// ---------------------------------------------------------------- fuse + scan
__global__ __launch_bounds__(256)
void k_fuse_scan(const float* __restrict__ x1, const float* __restrict__ x2,
                 float* __restrict__ xs) {
  int t = blockIdx.x * blockDim.x + threadIdx.x;
  const int total = BB * DD * HH * WW;
  if (t >= total) return;
  int w = t % WW;
  int h = (t / WW) % HH;
  int d = (t / (WW * HH)) % DD;
  int b = t / (WW * HH * DD);
  float a = x1[t], c = x2[t];
  float v = a * c + a + c;
  int he = h & 1, we = w & 1;
  int k = he + 2 * we;
  int l = he ? ((w >> 1) * W2 + (h >> 1)) : ((h >> 1) * W2 + (w >> 1));
  xs[(((size_t)(b * KK + k) * DD + d) * LL) + l] = v;
}

// --------------------------------------------------- x_proj GEMM (WMMA f32)
// per (b,k): out[c,l] = sum_d W[k][c][d] * xs[d,l];  M=44(->48), K=192, N=2304
// Row guard handled by clamped address + multiplicative mask: no exec-mask
// branches in the inner loop, unconditional b64 A-fragment loads.
__global__ __launch_bounds__(256)
void k_xproj_wmma(const float* __restrict__ xs, const float* __restrict__ xpw,
                  float* __restrict__ xdbl) {
  const int bk = blockIdx.z;
  const int k = bk & 3;
  const int m0 = blockIdx.y * 16;
  const int lane = threadIdx.x & 31;
  const int wid = threadIdx.x >> 5;
  const int n0 = blockIdx.x * 128 + wid * 16;
  const int ln = lane & 15;
  const int koff = (lane >= 16) ? 2 : 0;
  const int arow = m0 + ln;
  const int srow = (arow < CPROJ) ? arow : 0;        // clamped, loop-invariant
  const float mask = (arow < CPROJ) ? 1.f : 0.f;     // zero padded rows
  const float* abase = xpw + (size_t)k * CPROJ * DD + (size_t)srow * DD + koff;
  const float* bbase = xs + (size_t)bk * DD * LL + n0 + ln;
  v8f acc = {};
  for (int kk = 0; kk < DD; kk += 4) {
    v2f a = *(const v2f*)(abase + kk);               // contiguous, 8B aligned
    a.x *= mask;
    a.y *= mask;
    v2f bf;
    bf.x = bbase[(size_t)(kk + koff) * LL];
    bf.y = bbase[(size_t)(kk + koff + 1) * LL];
    acc = __builtin_amdgcn_wmma_f32_16x16x4_f32(false, a, false, bf,
                                                (short)0, acc, false, false);
  }
  float* obase = xdbl + (size_t)bk * CPAD * LL;
  int rbase = m0 + ((lane >= 16) ? 8 : 0);
#pragma unroll
  for (int j = 0; j < 8; ++j)
    obase[(size_t)(rbase + j) * LL + n0 + ln] = acc[j];
}

// ------------------------------------------------------ dt GEMM (WMMA f32)
// per (b,k): delta[d,l] = sum_r dtw[k][d][r] * xdbl[r,l]; M=192, K=12, N=2304
__global__ __launch_bounds__(256)
void k_dt_wmma(const float* __restrict__ xdbl, const float* __restrict__ dtw,
               float* __restrict__ delta) {
  const int bk = blockIdx.z;
  const int k = bk & 3;
  const int m0 = blockIdx.y * 16;
  const int lane = threadIdx.x & 31;
  const int wid = threadIdx.x >> 5;
  const int n0 = blockIdx.x * 128 + wid * 16;
  const int ln = lane & 15;
  const int koff = (lane >= 16) ? 2 : 0;
  const int arow = m0 + ln;
  const float* abase = dtw + ((size_t)k * DD + arow) * RR + koff;
  const float* bbase = xdbl + (size_t)bk * CPAD * LL + n0 + ln;
  v8f acc = {};
#pragma unroll
  for (int kk = 0; kk < RR; kk += 4) {
    v2f a = *(const v2f*)(abase + kk);               // 8B aligned (RR even)
    v2f bf;
    bf.x = bbase[(size_t)(kk + koff) * LL];
    bf.y = bbase[(size_t)(kk + koff + 1) * LL];
    acc = __builtin_amdgcn_wmma_f32_16x16x4_f32(false, a, false, bf,
                                                (short)0, acc, false, false);
  }
  float* obase = delta + (size_t)bk * DD * LL;
  int rbase = m0 + ((lane >= 16) ? 8 : 0);
#pragma unroll
  for (int j = 0; j < 8; ++j)
    obase[(size_t)(rbase + j) * LL + n0 + ln] = acc[j];
}

// ---------------------------------------------------------------- scan pass 1
// one block = (b,k,chunk); thread d holds 16 states; emits (prod a, local scan)
__global__ __launch_bounds__(192)
void k_scan_pass1(const float* __restrict__ xs, const float* __restrict__ xdbl,
                  const float* __restrict__ delta,
                  const float* __restrict__ A_logs,
                  const float* __restrict__ dt_bias,
                  float* __restrict__ aprod, float* __restrict__ bstate) {
  __shared__ float su[DD * LDP];
  __shared__ float sdt[DD * LDP];
  __shared__ float sB[NN * LDP];
  int bk = blockIdx.x / CHUNKS;
  int chunk = blockIdx.x % CHUNKS;
  int k = bk & 3;
  int d = threadIdx.x;
  int l0 = chunk * CLEN;
  float Aval[NN];
#pragma unroll
  for (int n = 0; n < NN; ++n)
    Aval[n] = -__expf(A_logs[(size_t)(k * DD + d) * NN + n]);
  float bias = dt_bias[k * DD + d];
  float p[NN], hb[NN];
#pragma unroll
  for (int n = 0; n < NN; ++n) { p[n] = 1.f; hb[n] = 0.f; }
  const float* ubase = xs + (size_t)bk * DD * LL + (size_t)d * LL;
  const float* dbase = delta + (size_t)bk * DD * LL + (size_t)d * LL;
  const float* Bbase = xdbl + ((size_t)bk * CPAD + RR) * LL;

  for (int ts = 0; ts < CLEN; ts += TSUB) {
    __syncthreads();
    int lt = l0 + ts;
    const float4* ur = (const float4*)(ubase + lt);
    const float4* dr = (const float4*)(dbase + lt);
#pragma unroll
    for (int c4 = 0; c4 < TSUB / 4; ++c4) {
      float4 uv = ur[c4], dv = dr[c4];
      su[d * LDP + c4 * 4 + 0] = uv.x;  su[d * LDP + c4 * 4 + 1] = uv.y;
      su[d * LDP + c4 * 4 + 2] = uv.z;  su[d * LDP + c4 * 4 + 3] = uv.w;
      sdt[d * LDP + c4 * 4 + 0] = dv.x; sdt[d * LDP + c4 * 4 + 1] = dv.y;
      sdt[d * LDP + c4 * 4 + 2] = dv.z; sdt[d * LDP + c4 * 4 + 3] = dv.w;
    }
    for (int idx = d; idx < NN * TSUB; idx += DD)
      sB[(idx / TSUB) * LDP + (idx % TSUB)] =
          Bbase[(size_t)(idx / TSUB) * LL + lt + (idx % TSUB)];
    __builtin_prefetch(ubase + lt + TSUB, 0, 1);
    __builtin_prefetch(dbase + lt + TSUB, 0, 1);
    __syncthreads();
#pragma unroll 4
    for (int s = 0; s < TSUB; ++s) {
      float dtr = sdt[d * LDP + s] + bias;
      float dt = dtr > 20.f ? dtr : log1pf(__expf(dtr));
      float du = dt * su[d * LDP + s];
#pragma unroll
      for (int n = 0; n < NN; ++n) {
        float a = __expf(dt * Aval[n]);
        p[n] *= a;
        hb[n] = hb[n] * a + du * sB[n * LDP + s];
      }
    }
  }
  size_t off = ((size_t)chunk * NCH + (size_t)bk * DD + d) * NN;
#pragma unroll
  for (int n = 0; n < NN; ++n) { aprod[off + n] = p[n]; bstate[off + n] = hb[n]; }
}

// ------------------------------------------------------------- chunk combine
__global__ __launch_bounds__(256)
void k_combine(const float* __restrict__ aprod, const float* __restrict__ bstate,
               float* __restrict__ hinit) {
  int t = blockIdx.x * blockDim.x + threadIdx.x;
  if (t >= NCH * NN) return;
  float h = 0.f;
  const size_t stride = (size_t)NCH * NN;
  for (int c = 0; c < CHUNKS; ++c) {
    size_t idx = (size_t)c * stride + t;
    hinit[idx] = h;
    h = aprod[idx] * h + bstate[idx];
  }
}

// ---------------------------------------------------------------- scan pass 2
__global__ __launch_bounds__(192)
void k_scan_pass2(const float* __restrict__ xs, const float* __restrict__ xdbl,
                  const float* __restrict__ delta,
                  const float* __restrict__ A_logs,
                  const float* __restrict__ dt_bias,
                  const float* __restrict__ Ds,
                  const float* __restrict__ hinit, float* __restrict__ yt) {
  __shared__ float su[DD * LDP];
  __shared__ float sdt[DD * LDP];
  __shared__ float sB[NN * LDP];
  __shared__ float sC[NN * LDP];
  int bk = blockIdx.x / CHUNKS;
  int chunk = blockIdx.x % CHUNKS;
  int k = bk & 3;
  int d = threadIdx.x;
  int l0 = chunk * CLEN;
  float Aval[NN];
#pragma unroll
  for (int n = 0; n < NN; ++n)
    Aval[n] = -__expf(A_logs[(size_t)(k * DD + d) * NN + n]);
  float bias = dt_bias[k * DD + d];
  float Dv = Ds[k * DD + d];
  float h[NN];
  size_t hoff = ((size_t)chunk * NCH + (size_t)bk * DD + d) * NN;
#pragma unroll
  for (int n = 0; n < NN; ++n) h[n] = hinit[hoff + n];
  const float* ubase = xs + (size_t)bk * DD * LL + (size_t)d * LL;
  const float* dbase = delta + (size_t)bk * DD * LL + (size_t)d * LL;
  const float* Bbase = xdbl + ((size_t)bk * CPAD + RR) * LL;
  const float* Cbase = xdbl + ((size_t)bk * CPAD + RR + NN) * LL;
  float* ybase = yt + (size_t)bk * LL * DD;

  for (int ts = 0; ts < CLEN; ts += TSUB) {
    __syncthreads();
    int lt = l0 + ts;
    const float4* ur = (const float4*)(ubase + lt);
    const float4* dr = (const float4*)(dbase + lt);
#pragma unroll
    for (int c4 = 0; c4 < TSUB / 4; ++c4) {
      float4 uv = ur[c4], dv = dr[c4];
      su[d * LDP + c4 * 4 + 0] = uv.x;  su[d * LDP + c4 * 4 + 1] = uv.y;
      su[d * LDP + c4 * 4 + 2] = uv.z;  su[d * LDP + c4 * 4 + 3] = uv.w;
      sdt[d * LDP + c4 * 4 + 0] = dv.x; sdt[d * LDP + c4 * 4 + 1] = dv.y;
      sdt[d * LDP + c4 * 4 + 2] = dv.z; sdt[d * LDP + c4 * 4 + 3] = dv.w;
    }
    for (int idx = d; idx < NN * TSUB; idx += DD) {
      int rr = idx / TSUB, cc = idx % TSUB;
      sB[rr * LDP + cc] = Bbase[(size_t)rr * LL + lt + cc];
      sC[rr * LDP + cc] = Cbase[(size_t)rr * LL + lt + cc];
    }
    __builtin_prefetch(ubase + lt + TSUB, 0, 1);
    __builtin_prefetch(dbase + lt + TSUB, 0, 1);
    __syncthreads();
#pragma unroll 4
    for (int s = 0; s < TSUB; ++s) {
      float dtr = sdt[d * LDP + s] + bias;
      float dt = dtr > 20.f ? dtr : log1pf(__expf(dtr));
      float uv = su[d * LDP + s];
      float du = dt * uv;
      float y = 0.f;
#pragma unroll
      for (int n = 0; n < NN; ++n) {
        float a = __expf(dt * Aval[n]);
        h[n] = h[n] * a + du * sB[n * LDP + s];
        y += h[n] * sC[n * LDP + s];
      }
      ybase[(size_t)(lt + s) * DD + d] = y + uv * Dv;  // coalesced over d
    }
  }
}

// --------------------------------------------------------- merge + LayerNorm
__global__ __launch_bounds__(256)
void k_merge_ln(const float* __restrict__ yt, const float* __restrict__ gw,
                const float* __restrict__ gb, float* __restrict__ out) {
  int wid = threadIdx.x >> 5;
  int lane = threadIdx.x & 31;
  int pix = blockIdx.x * 8 + wid;        // exact grid: B*H*W/8 blocks
  int w = pix % WW;
  int h = (pix / WW) % HH;
  int b = pix / (WW * HH);
  int he = h & 1, we = w & 1;
  int k = he + 2 * we;
  int l = he ? ((w >> 1) * W2 + (h >> 1)) : ((h >> 1) * W2 + (w >> 1));
  const float* src = yt + ((size_t)(b * KK + k) * LL + l) * DD;
  float v[6];
  float s = 0.f, s2 = 0.f;
#pragma unroll
  for (int i = 0; i < 6; ++i) {
    float x = src[lane + i * 32];
    v[i] = x; s += x; s2 += x * x;
  }
#pragma unroll
  for (int o = 16; o >= 1; o >>= 1) {
    s += __shfl_xor(s, o, 32);
    s2 += __shfl_xor(s2, o, 32);
  }
  float mean = s * (1.f / DD);
  float var = s2 * (1.f / DD) - mean * mean;
  float rstd = rsqrtf(var + 1e-5f);
  float* dst = out + (size_t)pix * DD;
#pragma unroll
  for (int i = 0; i < 6; ++i) {
    int dch = lane + i * 32;
    dst[dch] = (v[i] - mean) * rstd * gw[dch] + gb[dch];
  }
}

// ----------------------------------------------------------------------------
extern "C" void kernel_launch(void* const* d_in, const int* in_sizes, int n_in,
                              void* d_out, int out_size, void* d_ws,
                              size_t ws_size, hipStream_t stream) {
  (void)in_sizes; (void)n_in; (void)out_size; (void)ws_size;
  const float* x1   = (const float*)d_in[0];
  const float* x2   = (const float*)d_in[1];
  const float* xpw  = (const float*)d_in[2];  // (K,44,D)
  const float* dtw  = (const float*)d_in[3];  // (K,D,R)
  const float* dtb  = (const float*)d_in[4];  // (K,D)
  const float* Alog = (const float*)d_in[5];  // (K*D,N)
  const float* Ds   = (const float*)d_in[6];  // (K*D,)
  const float* gw   = (const float*)d_in[7];
  const float* gb   = (const float*)d_in[8];
  float* out = (float*)d_out;

  float* ws = (float*)d_ws;
  float* xs     = ws;                                      // 16*192*2304
  float* xdbl   = xs + (size_t)BKTOT * DD * LL;            // 16*48*2304
  float* delta  = xdbl + (size_t)BKTOT * CPAD * LL;        // 16*192*2304
  float* aprod  = delta + (size_t)BKTOT * DD * LL;         // 36*3072*16
  float* bstate = aprod + (size_t)CHUNKS * NCH * NN;
  float* hinit  = bstate + (size_t)CHUNKS * NCH * NN;
  float* yt     = hinit + (size_t)CHUNKS * NCH * NN;       // 16*2304*192

  const int totalX = BB * DD * HH * WW;
  k_fuse_scan<<<(totalX + 255) / 256, 256, 0, stream>>>(x1, x2, xs);

  dim3 g2(LL / 128, CPAD / 16, BKTOT);  // 18 x 3 x 16
  k_xproj_wmma<<<g2, 256, 0, stream>>>(xs, xpw, xdbl);

  dim3 g3(LL / 128, DD / 16, BKTOT);    // 18 x 12 x 16
  k_dt_wmma<<<g3, 256, 0, stream>>>(xdbl, dtw, delta);

  k_scan_pass1<<<BKTOT * CHUNKS, DD, 0, stream>>>(xs, xdbl, delta, Alog, dtb,
                                                  aprod, bstate);
  k_combine<<<(NCH * NN + 255) / 256, 256, 0, stream>>>(aprod, bstate, hinit);
  k_scan_pass2<<<BKTOT * CHUNKS, DD, 0, stream>>>(xs, xdbl, delta, Alog, dtb,
                                                  Ds, hinit, yt);
  k_merge_ln<<<(BB * HH * WW) / 8, 256, 0, stream>>>(yt, gw, gb, out);
}